// SSDChunkwiseScan_5909874999633
// MI455X (gfx1250) — compile-verified
//
#include <hip/hip_runtime.h>

typedef __attribute__((ext_vector_type(2))) float v2f;
typedef __attribute__((ext_vector_type(8))) float v8f;

// Problem dimensions from the reference setup_inputs()
constexpr int B_ = 2, L_ = 8192, H_ = 16, D_ = 64, C_ = 64, NC_ = L_ / C_;

// ---------------------------------------------------------------------------
// Pass 1: intra-chunk scan. One wave32 per (b, chunk, h). Lane owns columns
// d = 2*lane, 2*lane+1. State h[2][2-cols] and cumulative 2x2 product P live
// in registers. Writes local_h into Y, cum_A/total_A/final_h into workspace.
// ---------------------------------------------------------------------------
__global__ __launch_bounds__(256) void intra_chunk_kernel(
    const float* __restrict__ A, const float* __restrict__ K,
    const float* __restrict__ V, const float* __restrict__ beta,
    float* __restrict__ Y, float* __restrict__ cumA,
    float* __restrict__ totalA, float* __restrict__ finalH)
{
  const int lane = threadIdx.x & 31;
  const int wave = (blockIdx.x * blockDim.x + threadIdx.x) >> 5;
  const int h = wave % H_;
  const int n = (wave / H_) % NC_;
  const int b = wave / (H_ * NC_);
  const int l0 = n * C_;

  const long baseLH = (long)(b * L_ + l0) * H_ + h;
  const float* aP = A    + baseLH * 4;
  const float* kP = K    + baseLH * D_ + 2 * lane;
  const float* vP = V    + baseLH * 2;
  const float* bP = beta + baseLH;
  float* yP = Y    + baseLH * (2 * D_) + 2 * lane;
  float* cP = cumA + baseLH * 4;

  float h00 = 0.f, h01 = 0.f, h10 = 0.f, h11 = 0.f;   // h[i][c], c = my 2 cols
  float P00 = 1.f, P01 = 0.f, P10 = 0.f, P11 = 1.f;   // cumulative 2x2 product

  for (int t = 0; t < C_; ++t) {
    const float4 a = *(const float4*)(aP + (long)t * (H_ * 4));
    const float2 k = *(const float2*)(kP + (long)t * (H_ * D_));
    const float2 v = *(const float2*)(vP + (long)t * (H_ * 2));
    const float bt = bP[(long)t * H_];

    // h_prop = a @ h
    const float hp00 = a.x * h00 + a.y * h10;
    const float hp01 = a.x * h01 + a.y * h11;
    const float hp10 = a.z * h00 + a.w * h10;
    const float hp11 = a.z * h01 + a.w * h11;

    // read_i = sum_d h_prop[i,d] * k[d]  (wave32 butterfly reduction)
    float r0 = hp00 * k.x + hp01 * k.y;
    float r1 = hp10 * k.x + hp11 * k.y;
    #pragma unroll
    for (int m = 16; m >= 1; m >>= 1) {
      r0 += __shfl_xor(r0, m, 32);
      r1 += __shfl_xor(r1, m, 32);
    }

    const float d0 = bt * (v.x - r0);
    const float d1 = bt * (v.y - r1);
    h00 = hp00 + d0 * k.x;  h01 = hp01 + d0 * k.y;
    h10 = hp10 + d1 * k.x;  h11 = hp11 + d1 * k.y;

    // local_h -> Y (pass 3 adds the correction in place)
    float* y = yP + (long)t * (H_ * 2 * D_);
    *(float2*)(y)      = make_float2(h00, h01);
    *(float2*)(y + D_) = make_float2(h10, h11);

    // P_new = a @ P (uniform per wave; lane 0 stores cum_A)
    const float nP00 = a.x * P00 + a.y * P10;
    const float nP01 = a.x * P01 + a.y * P11;
    const float nP10 = a.z * P00 + a.w * P10;
    const float nP11 = a.z * P01 + a.w * P11;
    P00 = nP00; P01 = nP01; P10 = nP10; P11 = nP11;
    if (lane == 0)
      *(float4*)(cP + (long)t * (H_ * 4)) = make_float4(P00, P01, P10, P11);
  }

  const long cnh = (long)(b * NC_ + n) * H_ + h;
  if (lane == 0)
    *(float4*)(totalA + cnh * 4) = make_float4(P00, P01, P10, P11);
  float* fh = finalH + cnh * (2 * D_) + 2 * lane;
  *(float2*)(fh)      = make_float2(h00, h01);
  *(float2*)(fh + D_) = make_float2(h10, h11);
}

// ---------------------------------------------------------------------------
// Pass 2: inter-chunk exclusive scan. One block per (b,h); thread d owns one
// state column. states[n] = state BEFORE chunk n is applied (exclusive).
// ---------------------------------------------------------------------------
__global__ __launch_bounds__(64) void inter_chunk_kernel(
    const float* __restrict__ totalA, const float* __restrict__ finalH,
    float* __restrict__ states)
{
  const int d = threadIdx.x;
  const int h = blockIdx.x % H_;
  const int b = blockIdx.x / H_;
  float s0 = 0.f, s1 = 0.f;
  for (int n = 0; n < NC_; ++n) {
    const long idx = (long)(b * NC_ + n) * H_ + h;
    states[idx * 2 * D_ + d]      = s0;
    states[idx * 2 * D_ + D_ + d] = s1;
    const float4 a = *(const float4*)(totalA + idx * 4);
    const float f0 = finalH[idx * 2 * D_ + d];
    const float f1 = finalH[idx * 2 * D_ + D_ + d];
    const float ns0 = a.x * s0 + a.y * s1 + f0;
    const float ns1 = a.z * s0 + a.w * s1 + f1;
    s0 = ns0; s1 = ns1;
  }
}

// ---------------------------------------------------------------------------
// Pass 3: Y += cum_A @ chunk_states via V_WMMA_F32_16X16X4_F32.
// Per (b, chunk, h, i): M=64 timesteps, N=64 state cols, K=2 (padded to 4
// with zeroed lanes 16-31, which hold K=2,3 per the 32-bit A/B layouts).
// One block of 4 waves; wave w handles timestep rows t0 = 16*w, looping the
// four 16-wide column tiles. C/D use the documented 8-VGPR 16x16 layout:
// element (VGPR r, lane) = row t0 + r + (lane>=16 ? 8 : 0), col = lane&15.
// ---------------------------------------------------------------------------
__global__ __launch_bounds__(128) void correction_wmma_kernel(
    const float* __restrict__ cumA, const float* __restrict__ states,
    float* __restrict__ Y)
{
  const int lane = threadIdx.x & 31;
  const int wid  = threadIdx.x >> 5;
  int bid = blockIdx.x;
  const int i = bid & 1;  bid >>= 1;
  const int h = bid % H_; bid /= H_;
  const int n = bid % NC_;
  const int b = bid / NC_;
  const int t0 = wid * 16;

  const int  lo     = lane & 15;
  const bool hiHalf = lane >= 16;
  const int  rOff   = hiHalf ? 8 : 0;

  // A-tile: lanes 0-15 carry (K=0,K=1) = (cumA[t,i,0], cumA[t,i,1]); K=2,3 = 0
  const float* cA = cumA + (((long)(b * L_ + n * C_ + t0) * H_ + h) * 2 + i) * 2;
  v2f amat = {0.f, 0.f};
  if (!hiHalf) {
    const float2 p = *(const float2*)(cA + (long)lo * (H_ * 4));
    amat[0] = p.x; amat[1] = p.y;
  }

  const float* S = states + ((long)(b * NC_ + n) * H_ + h) * (2 * D_);
  float* yBase = Y + (((long)(b * L_ + n * C_ + t0) * H_ + h) * 2 + i) * D_;
  const long tStride = (long)H_ * 2 * D_;   // Y stride between timesteps

  #pragma unroll
  for (int d0 = 0; d0 < D_; d0 += 16) {
    // B-tile: lanes 0-15 carry (K=0,K=1) = (S[0,d], S[1,d]); K=2,3 = 0
    v2f bmat = {0.f, 0.f};
    if (!hiHalf) {
      bmat[0] = S[d0 + lo];
      bmat[1] = S[D_ + d0 + lo];
    }
    v8f c;
    #pragma unroll
    for (int r = 0; r < 8; ++r)
      c[r] = yBase[(long)(r + rOff) * tStride + d0 + lo];

    c = __builtin_amdgcn_wmma_f32_16x16x4_f32(false, amat, false, bmat,
                                              (short)0, c, false, false);

    #pragma unroll
    for (int r = 0; r < 8; ++r)
      yBase[(long)(r + rOff) * tStride + d0 + lo] = c[r];
  }
}

// ---------------------------------------------------------------------------
extern "C" void kernel_launch(void* const* d_in, const int* in_sizes, int n_in,
                              void* d_out, int out_size, void* d_ws, size_t ws_size,
                              hipStream_t stream)
{
  const float* A    = (const float*)d_in[0];   // (B,L,H,2,2)
  const float* K    = (const float*)d_in[1];   // (B,L,H,D)
  const float* V    = (const float*)d_in[2];   // (B,L,H,2)
  const float* beta = (const float*)d_in[3];   // (B,L,H)

  float* Y      = (float*)d_out;                        // (B,L,H,2,D)
  float* states = Y + (size_t)B_ * L_ * H_ * 2 * D_;    // (B,NC,H,2,D)

  char* ws = (char*)d_ws;
  float* cumA   = (float*)ws;                                            // B*L*H*4
  float* totalA = (float*)(ws + (size_t)B_ * L_ * H_ * 4 * sizeof(float));      // B*NC*H*4
  float* finalH = (float*)((char*)totalA + (size_t)B_ * NC_ * H_ * 4 * sizeof(float)); // B*NC*H*2*D

  // Pass 1: 4096 waves, 8 waves / block
  intra_chunk_kernel<<<dim3((B_ * NC_ * H_) / 8), dim3(256), 0, stream>>>(
      A, K, V, beta, Y, cumA, totalA, finalH);

  // Pass 2: one block per (b,h)
  inter_chunk_kernel<<<dim3(B_ * H_), dim3(64), 0, stream>>>(totalA, finalH, states);

  // Pass 3: one block per (b,chunk,h,i), 4 waves each
  correction_wmma_kernel<<<dim3(B_ * NC_ * H_ * 2), dim3(128), 0, stream>>>(
      cumA, states, Y);
}